// ArtemisManualFeatures_81853486727371
// MI455X (gfx1250) — compile-verified
//
#include <hip/hip_runtime.h>
#include <hip/hip_bf16.h>

typedef float v2f __attribute__((ext_vector_type(2)));
typedef float v4f __attribute__((ext_vector_type(4)));
typedef float v8f __attribute__((ext_vector_type(8)));

#define THREADS 256

__device__ __forceinline__ float wave_sum(float v) {
#pragma unroll
    for (int off = 16; off > 0; off >>= 1) v += __shfl_down(v, off, 32);
    return v;
}
__device__ __forceinline__ float wave_minf(float v) {
#pragma unroll
    for (int off = 16; off > 0; off >>= 1) v = fminf(v, __shfl_down(v, off, 32));
    return v;
}
__device__ __forceinline__ float wave_maxf(float v) {
#pragma unroll
    for (int off = 16; off > 0; off >>= 1) v = fmaxf(v, __shfl_down(v, off, 32));
    return v;
}

// ---------------------------------------------------------------------------
// Kernel 1: per-row streaming reductions. One block (8 wave32) per row.
// Writes padded feature matrix feats[B][32] (cols 29..31 = 0).
// ---------------------------------------------------------------------------
__global__ __launch_bounds__(THREADS)
void feat_kernel(const float* __restrict__ prices, const float* __restrict__ volumes,
                 const float* __restrict__ holding, const float* __restrict__ ua,
                 const float* __restrict__ tc, const float* __restrict__ cc,
                 float* __restrict__ feats, int T) {
    const int row = blockIdx.x;
    const int tid = threadIdx.x;
    const size_t base = (size_t)row * (size_t)T;
    const v4f* pr = (const v4f*)(prices + base);
    const v4f* vo = (const v4f*)(volumes + base);
    const v4f* ho = (const v4f*)(holding + base);
    const int nvec = T >> 2;

    // 6 bits per bin; per-thread element count = T/THREADS = 32 <= 63.
    unsigned long long bpack = 0ull;  // benford bins 0..8
    unsigned long long rpack = 0ull;  // rounding bins 0..9
    float hsum = 0.f, hsq = 0.f, vsum = 0.f, vsq = 0.f;
    float hmin = 3.402823466e38f, hmax = -3.402823466e38f;

    for (int i = tid; i < nvec; i += THREADS) {
        v4f p = __builtin_nontemporal_load(pr + i);
        v4f v = __builtin_nontemporal_load(vo + i);
        v4f h = __builtin_nontemporal_load(ho + i);
#pragma unroll
        for (int c = 0; c < 4; ++c) {
            float av = floorf(fabsf(p[c]));
            unsigned iv = (unsigned)av;
            unsigned last = iv % 10u;
            rpack += 1ull << (6u * last);
            // exact power-of-10 scale (all exactly representable in f32)
            float s = 1.0f;
            s = (av >= 10.f) ? 10.f : s;
            s = (av >= 100.f) ? 100.f : s;
            s = (av >= 1000.f) ? 1000.f : s;
            s = (av >= 10000.f) ? 10000.f : s;
            s = (av >= 100000.f) ? 100000.f : s;
            s = (av >= 1000000.f) ? 1000000.f : s;
            s = (av >= 10000000.f) ? 10000000.f : s;
            int fd = (int)(av / s);           // correctly-rounded f32 divide
            fd = fd < 1 ? 1 : (fd > 9 ? 9 : fd);
            bpack += 1ull << (6u * (unsigned)(fd - 1));

            float hv = h[c];
            hsum += hv; hsq += hv * hv;
            hmin = fminf(hmin, hv); hmax = fmaxf(hmax, hv);
            float vv = v[c];
            vsum += vv; vsq += vv * vv;
        }
    }

    __shared__ unsigned s_hist[19];
    __shared__ float s_acc[4];            // hsum, hsq, vsum, vsq
    __shared__ unsigned s_min, s_max;     // uint-punned non-negative floats
    if (tid < 19) s_hist[tid] = 0u;
    if (tid < 4)  s_acc[tid] = 0.f;
    if (tid == 0) { s_min = 0x7F800000u; s_max = 0u; }
    __syncthreads();

#pragma unroll
    for (int b = 0; b < 9; ++b) {
        unsigned cnt = (unsigned)((bpack >> (6 * b)) & 63ull);
        if (cnt) atomicAdd(&s_hist[b], cnt);
    }
#pragma unroll
    for (int b = 0; b < 10; ++b) {
        unsigned cnt = (unsigned)((rpack >> (6 * b)) & 63ull);
        if (cnt) atomicAdd(&s_hist[9 + b], cnt);
    }
    hsum = wave_sum(hsum); hsq = wave_sum(hsq);
    vsum = wave_sum(vsum); vsq = wave_sum(vsq);
    hmin = wave_minf(hmin); hmax = wave_maxf(hmax);
    if ((tid & 31) == 0) {
        atomicAdd(&s_acc[0], hsum); atomicAdd(&s_acc[1], hsq);
        atomicAdd(&s_acc[2], vsum); atomicAdd(&s_acc[3], vsq);
        atomicMin(&s_min, __float_as_uint(hmin));
        atomicMax(&s_max, __float_as_uint(hmax));
    }
    __syncthreads();

    if (tid < 32) {
        const float Tf = (float)T;
        float outv = 0.f;
        if (tid < 9) {
            float d = (float)(tid + 1);
            float expected = log10f((d + 1.f) / d);
            outv = fabsf((float)s_hist[tid] / Tf - expected);
        } else if (tid < 19) {
            outv = (float)s_hist[tid] / Tf;
        } else if (tid == 19) outv = s_acc[0] / Tf;
        else if (tid == 20) outv = __uint_as_float(s_min);
        else if (tid == 21) outv = __uint_as_float(s_max);
        else if (tid == 22) outv = sqrtf(fmaxf(0.f, (s_acc[1] - s_acc[0] * s_acc[0] / Tf) / (Tf - 1.f)));
        else if (tid == 23) outv = ua[row];
        else if (tid == 24) outv = tc[row];
        else if (tid == 25) outv = cc[row];
        else if (tid == 26) outv = s_acc[2] / Tf;
        else if (tid == 27) outv = sqrtf(fmaxf(0.f, (s_acc[3] - s_acc[2] * s_acc[2] / Tf) / (Tf - 1.f)));
        else if (tid == 28) outv = s_acc[2];
        feats[(size_t)row * 32 + tid] = outv;
    }
}

// ---------------------------------------------------------------------------
// Kernel 2: out[B,32] = feats[B,32] @ Wcomb^T + bias via V_WMMA_F32_16X16X4_F32.
// 8 waves/block, 16 rows per wave, 2 N-tiles, 8 K-steps of 4 => 16 WMMA/wave.
// ---------------------------------------------------------------------------
__global__ __launch_bounds__(THREADS)
void proj_kernel(const float* __restrict__ feats,
                 const float* __restrict__ Wb, const float* __restrict__ bb,
                 const float* __restrict__ Wr, const float* __restrict__ br,
                 const float* __restrict__ Wt, const float* __restrict__ bt,
                 const float* __restrict__ Wa, const float* __restrict__ ba,
                 float* __restrict__ out, int B) {
    __shared__ float sW[32 * 32];   // sW[k*32 + n] = Wcomb[n][k], K-major
    __shared__ float sBias[32];
    const int tid = threadIdx.x;

    for (int idx = tid; idx < 1024; idx += THREADS) {
        int k = idx >> 5, n = idx & 31;
        float w = 0.f;
        if (n < 8)       { if (k < 9) w = Wb[n * 9 + k]; }
        else if (n < 16) { int kk = k - 9;  if (kk >= 0 && kk < 10) w = Wr[(n - 8) * 10 + kk]; }
        else if (n < 24) { int kk = k - 19; if (kk >= 0 && kk < 4)  w = Wt[(n - 16) * 4 + kk]; }
        else             { int kk = k - 23; if (kk >= 0 && kk < 6)  w = Wa[(n - 24) * 6 + kk]; }
        sW[idx] = w;
    }
    if (tid < 32) {
        float b;
        if (tid < 8)       b = bb[tid];
        else if (tid < 16) b = br[tid - 8];
        else if (tid < 24) b = bt[tid - 16];
        else               b = ba[tid - 24];
        sBias[tid] = b;
    }
    __syncthreads();

    const int lane  = tid & 31;
    const int wave  = tid >> 5;
    const int m0    = blockIdx.x * 128 + wave * 16;
    const int nl    = lane & 15;
    const int khalf = (lane >> 4) * 2;             // lanes 0-15: K+0/K+1, lanes 16-31: K+2/K+3
    const int mrow  = m0 + nl;
    const int mld   = (mrow < B) ? mrow : (B - 1); // clamp: keep EXEC all-ones for WMMA

    v8f c0 = {0.f, 0.f, 0.f, 0.f, 0.f, 0.f, 0.f, 0.f};
    v8f c1 = {0.f, 0.f, 0.f, 0.f, 0.f, 0.f, 0.f, 0.f};

#pragma unroll
    for (int kt = 0; kt < 8; ++kt) {
        const int kk = kt * 4 + khalf;
        v2f a, b0, b1;
        a.x  = feats[(size_t)mld * 32 + kk];
        a.y  = feats[(size_t)mld * 32 + kk + 1];
        b0.x = sW[kk * 32 + nl];
        b0.y = sW[(kk + 1) * 32 + nl];
        b1.x = sW[kk * 32 + 16 + nl];
        b1.y = sW[(kk + 1) * 32 + 16 + nl];
        c0 = __builtin_amdgcn_wmma_f32_16x16x4_f32(false, a, false, b0, (short)0, c0, false, false);
        c1 = __builtin_amdgcn_wmma_f32_16x16x4_f32(false, a, false, b1, (short)0, c1, false, false);
    }

#pragma unroll
    for (int r = 0; r < 8; ++r) {
        int m = m0 + ((lane < 16) ? r : r + 8);
        if (m < B) {
            out[(size_t)m * 32 + nl]      = c0[r] + sBias[nl];
            out[(size_t)m * 32 + 16 + nl] = c1[r] + sBias[16 + nl];
        }
    }
}

extern "C" void kernel_launch(void* const* d_in, const int* in_sizes, int n_in,
                              void* d_out, int out_size, void* d_ws, size_t ws_size,
                              hipStream_t stream) {
    const float* prices  = (const float*)d_in[0];
    const float* volumes = (const float*)d_in[1];
    const float* holding = (const float*)d_in[2];
    const float* ua      = (const float*)d_in[3];
    const float* tc      = (const float*)d_in[4];
    const float* cc      = (const float*)d_in[5];
    const float* Wb      = (const float*)d_in[6];
    const float* bb      = (const float*)d_in[7];
    const float* Wr      = (const float*)d_in[8];
    const float* br      = (const float*)d_in[9];
    const float* Wt      = (const float*)d_in[10];
    const float* bt      = (const float*)d_in[11];
    const float* Wa      = (const float*)d_in[12];
    const float* ba      = (const float*)d_in[13];

    const int B = in_sizes[3];            // unique_addresses is [B]
    const int T = in_sizes[0] / B;

    float* feats = (float*)d_ws;          // B*32 floats scratch

    feat_kernel<<<B, THREADS, 0, stream>>>(prices, volumes, holding, ua, tc, cc, feats, T);

    const int blocks2 = (B + 127) / 128;  // 128 rows per block (8 waves x 16 rows)
    proj_kernel<<<blocks2, THREADS, 0, stream>>>(feats, Wb, bb, Wr, br, Wt, bt, Wa, ba,
                                                 (float*)d_out, B);
}